// IGAE_encoder_67070209294347
// MI455X (gfx1250) — compile-verified
//
#include <hip/hip_runtime.h>
#include <hip/hip_bf16.h>

typedef __attribute__((ext_vector_type(2))) float v2f;
typedef __attribute__((ext_vector_type(8))) float v8f;

static constexpr int NN = 8192;      // nodes
static constexpr int NE = 262144;    // edges
static constexpr float SLOPE = 0.2f;

// d_out layout (flat concatenation, floats):
// z_igae [8192,20] | z_adj [8192,8192] | az1 [8192,128] | az2 [8192,256]
// | az3 [8192,20] | z1 [8192,128] | z2 [8192,256]
static constexpr long long O_ZIGAE = 0;
static constexpr long long O_ADJ   = 163840;
static constexpr long long O_AZ1   = 67272704;
static constexpr long long O_AZ2   = 68321280;
static constexpr long long O_AZ3   = 70418432;
static constexpr long long O_Z1    = 70582272;
static constexpr long long O_Z2    = 71630848;

// ---------------------------------------------------------------- zero fill
__global__ __launch_bounds__(256) void zero_f4(float4* __restrict__ p, long long n4) {
  long long i = (long long)blockIdx.x * blockDim.x + threadIdx.x;
  if (i < n4) p[i] = make_float4(0.f, 0.f, 0.f, 0.f);
}

// ---------------------------------------------------------------- dense GEMM
// C[M,N] = act(A[M,K] @ B[K,N]) using V_WMMA_F32_16X16X4_F32.
// One wave -> 16 rows x (NTILES*16) cols. A fragment reused across NTILES.
// Fragment layout (32-bit, 16x4): lanes 0-15 hold K={0,1}, lanes 16-31 K={2,3};
// M (for A) / N (for B) striped across the 16 lanes of each half.
template <int NTILES, bool LEAKY>
__global__ __launch_bounds__(256) void gemm_wmma(const float* __restrict__ A,
                                                 const float* __restrict__ B,
                                                 float* __restrict__ C,
                                                 int M, int N, int K) {
  const int wave = threadIdx.x >> 5;
  const int lane = threadIdx.x & 31;
  const int half = lane >> 4;       // 0: K even pair, 1: K odd pair
  const int lm   = lane & 15;
  const int m0   = (blockIdx.x * 8 + wave) * 16;
  if (m0 >= M) return;              // wave-uniform: EXEC stays all-1s below

  v8f acc[NTILES];
#pragma unroll
  for (int j = 0; j < NTILES; ++j) acc[j] = v8f{0, 0, 0, 0, 0, 0, 0, 0};

  const float* arow = A + (long long)(m0 + lm) * K;
  for (int kb = 0; kb < K; kb += 4) {
    v2f a;
    a.x = arow[kb + 2 * half];
    a.y = arow[kb + 2 * half + 1];
    const float* b0 = B + (long long)(kb + 2 * half) * N;
#pragma unroll
    for (int j = 0; j < NTILES; ++j) {
      const int n = j * 16 + lm;
      v2f b;
      b.x = (n < N) ? b0[n] : 0.f;       // lane predication -> v_cndmask
      b.y = (n < N) ? b0[N + n] : 0.f;
      acc[j] = __builtin_amdgcn_wmma_f32_16x16x4_f32(
          false, a, false, b, (short)0, acc[j], false, false);
    }
  }

#pragma unroll
  for (int j = 0; j < NTILES; ++j) {
    const int n = j * 16 + lm;
    if (n < N) {
#pragma unroll
      for (int r = 0; r < 8; ++r) {       // VGPR r: rows m0+r (half 0), m0+8+r (half 1)
        float v = acc[j][r];
        if (LEAKY) v = (v >= 0.f) ? v : SLOPE * v;
        C[(long long)(m0 + r + 8 * half) * N + n] = v;
      }
    }
  }
}

// ---------------------------------------------------------------- SpMM
// out[rows[e], :] += vals[e] * feat[cols[e], :]   (atomic scatter-add)
template <int D>
__global__ __launch_bounds__(256) void spmm(const int* __restrict__ rows,
                                            const int* __restrict__ cols,
                                            const float* __restrict__ vals,
                                            const float* __restrict__ feat,
                                            float* __restrict__ out) {
  long long tid = (long long)blockIdx.x * blockDim.x + threadIdx.x;
  if (tid >= (long long)NE * D) return;
  const int e = (int)(tid / D);
  const int d = (int)(tid - (long long)e * D);
  const int r = rows[e];
  const int c = cols[e];
  atomicAdd(&out[(long long)r * D + d], vals[e] * feat[(long long)c * D + d]);
}

// ---------------------------------------------------------------- decoder
// adj = sigmoid(Z @ Z^T), Z: [NN,20]. One wave -> 16x64 strip (4 WMMA tiles,
// shared A fragment, K=20 fully unrolled).
__global__ __launch_bounds__(256) void decoder_wmma(const float* __restrict__ Z,
                                                    float* __restrict__ adj) {
  constexpr int DZ = 20;
  constexpr int NTJ = 4;                       // 4 col tiles per wave
  const int wave = threadIdx.x >> 5;
  const int lane = threadIdx.x & 31;
  const int half = lane >> 4;
  const int lm   = lane & 15;

  const int job = blockIdx.x * 8 + wave;       // NN/16 * NN/(16*NTJ) jobs
  const int ti  = job / (NN / (16 * NTJ));
  const int tjg = job % (NN / (16 * NTJ));
  const int m0  = ti * 16;
  const int n0  = tjg * (16 * NTJ);

  v8f acc[NTJ];
#pragma unroll
  for (int j = 0; j < NTJ; ++j) acc[j] = v8f{0, 0, 0, 0, 0, 0, 0, 0};

  const float* za = Z + (long long)(m0 + lm) * DZ;
#pragma unroll
  for (int kb = 0; kb < DZ; kb += 4) {
    v2f a;
    a.x = za[kb + 2 * half];
    a.y = za[kb + 2 * half + 1];
#pragma unroll
    for (int j = 0; j < NTJ; ++j) {
      const float* zb = Z + (long long)(n0 + j * 16 + lm) * DZ;  // B = Z^T
      v2f b;
      b.x = zb[kb + 2 * half];
      b.y = zb[kb + 2 * half + 1];
      acc[j] = __builtin_amdgcn_wmma_f32_16x16x4_f32(
          false, a, false, b, (short)0, acc[j], false, false);
    }
  }

#pragma unroll
  for (int j = 0; j < NTJ; ++j) {
#pragma unroll
    for (int r = 0; r < 8; ++r) {
      const float v = acc[j][r];
      adj[(long long)(m0 + r + 8 * half) * NN + n0 + j * 16 + lm] =
          1.f / (1.f + __expf(-v));
    }
  }
}

// ---------------------------------------------------------------- launch
extern "C" void kernel_launch(void* const* d_in, const int* in_sizes, int n_in,
                              void* d_out, int out_size, void* d_ws, size_t ws_size,
                              hipStream_t stream) {
  const float* x        = (const float*)d_in[0];
  const int*   adj_rows = (const int*)d_in[1];
  const int*   adj_cols = (const int*)d_in[2];
  const float* adj_vals = (const float*)d_in[3];
  const float* w1       = (const float*)d_in[4];
  const float* w2       = (const float*)d_in[5];
  const float* w3       = (const float*)d_in[6];

  float* out = (float*)d_out;
  float* ws  = (float*)d_ws;

  float* s1 = ws;                        // [8192,128]
  float* s2 = s1 + (long long)NN * 128;  // [8192,256]
  float* s3 = s2 + (long long)NN * 256;  // [8192,20]

  float* z_igae = out + O_ZIGAE;
  float* z_adj  = out + O_ADJ;
  float* az1    = out + O_AZ1;
  float* az2    = out + O_AZ2;
  float* az3    = out + O_AZ3;
  float* z1     = out + O_Z1;
  float* z2     = out + O_Z2;

  // Zero all scatter-add accumulators (z_igae and the contiguous az1..z2 tail).
  {
    long long n4a = (long long)NN * 20 / 4;                    // z_igae
    long long n4b = ((long long)73728000 - O_AZ1) / 4;         // az1..z2 block
    zero_f4<<<(int)((n4a + 255) / 256), 256, 0, stream>>>((float4*)z_igae, n4a);
    zero_f4<<<(int)((n4b + 255) / 256), 256, 0, stream>>>((float4*)az1, n4b);
  }

  const int gemm_grid = NN / (16 * 8);   // 64 blocks, 8 waves each

  // Layer 1: s1 = leaky(x @ w1); z1 = A s1; az1 = A z1
  gemm_wmma<8, true><<<gemm_grid, 256, 0, stream>>>(x, w1, s1, NN, 128, 3000);
  spmm<128><<<(int)(((long long)NE * 128 + 255) / 256), 256, 0, stream>>>(
      adj_rows, adj_cols, adj_vals, s1, z1);
  spmm<128><<<(int)(((long long)NE * 128 + 255) / 256), 256, 0, stream>>>(
      adj_rows, adj_cols, adj_vals, z1, az1);

  // Layer 2: s2 = leaky(z1 @ w2); z2 = A s2; az2 = A z2
  gemm_wmma<16, true><<<gemm_grid, 256, 0, stream>>>(z1, w2, s2, NN, 256, 128);
  spmm<256><<<(int)(((long long)NE * 256 + 255) / 256), 256, 0, stream>>>(
      adj_rows, adj_cols, adj_vals, s2, z2);
  spmm<256><<<(int)(((long long)NE * 256 + 255) / 256), 256, 0, stream>>>(
      adj_rows, adj_cols, adj_vals, z2, az2);

  // Layer 3 (no activation): s3 = z2 @ w3; z_igae = A s3; az3 = A z_igae
  gemm_wmma<2, false><<<gemm_grid, 256, 0, stream>>>(z2, w3, s3, NN, 20, 256);
  spmm<20><<<(int)(((long long)NE * 20 + 255) / 256), 256, 0, stream>>>(
      adj_rows, adj_cols, adj_vals, s3, z_igae);
  spmm<20><<<(int)(((long long)NE * 20 + 255) / 256), 256, 0, stream>>>(
      adj_rows, adj_cols, adj_vals, z_igae, az3);

  // Decoder: z_adj = sigmoid(z_igae @ z_igae^T)
  const int dec_jobs = (NN / 16) * (NN / 64);  // 512 * 128 waves
  decoder_wmma<<<dec_jobs / 8, 256, 0, stream>>>(z_igae, z_adj);
}